// ROIAlign_25589415149604
// MI455X (gfx1250) — compile-verified
//
#include <hip/hip_runtime.h>
#include <hip/hip_bf16.h>

typedef __attribute__((ext_vector_type(2))) float v2f;
typedef __attribute__((ext_vector_type(8))) float v8f;

#define C_TOTAL 256

__device__ __forceinline__ int iclamp(int v, int lo, int hi) {
    return v < lo ? lo : (v > hi ? hi : v);
}

// One workgroup (256 threads = 8 waves) per ROI.
// Separable bilinear crop as two WMMA GEMM stages, fp32 end-to-end.
//   stage 1: temp[py,xj,c] = sum_yi Wy[py,yi] * fm[yi, xj, c]     (V_WMMA_F32_16X16X4_F32)
//   stage 2: out[py,px,c]  = sum_xj Wx[px,xj] * temp[py,xj,c]     (V_WMMA_F32_16X16X4_F32)
__global__ __launch_bounds__(256) void roi_align_wmma_f32(
    const float* __restrict__ fm2, const float* __restrict__ fm3,
    const float* __restrict__ fm4, const float* __restrict__ fm5,
    const float* __restrict__ rois, float* __restrict__ out,
    int nroi_per_img)
{
    __shared__ float sWy[7][16];            // row-interp weights, K padded to 16
    __shared__ float sWx[7][16];            // col-interp weights, K padded to 16
    __shared__ int   sRows[16];             // gathered y indices (14 used, clamped pad)
    __shared__ int   sCols[16];             // gathered x indices
    __shared__ float sTemp[8][7][16][16];   // [wave][py][xj(pad16)][chan-in-block]

    const int blk  = blockIdx.x;
    const int b    = blk / nroi_per_img;
    const int roi  = blk - b * nroi_per_img;
    const int tid  = threadIdx.x;
    const int lane = tid & 31;
    const int wave = tid >> 5;

    const float* rp  = rois + (size_t)(b * nroi_per_img + roi) * 4;
    const float rx1 = rp[0], ry1 = rp[1], rx2 = rp[2], ry2 = rp[3];

    // ---- FPN level select; force scalar (uniform across block) so the fm
    //      base lands in SGPRs and gathers use saddr + 32-bit voffset ----
    const float area = (ry2 - ry1) * (rx2 - rx1);
    int lvl = (int)rintf(0.5f * log2f(area) - 7.8073549220576042f + 4.0f);
    lvl = iclamp(lvl, 2, 5);
    lvl = __builtin_amdgcn_readfirstlane(lvl);

    const int H = 1024 >> lvl;                          // W == H; stride*H == 1024 for all levels
    const float sc = (float)(H - 1) * (1.0f / 1024.0f); // ((v/stride)/H)*(H-1) == v*(H-1)/1024
    const float* fm = (lvl == 2) ? fm2 : (lvl == 3) ? fm3 : (lvl == 4) ? fm4 : fm5;
    fm += (unsigned)b * (unsigned)(H * H * C_TOTAL);    // <= 16.7M elements: 32-bit safe
    const unsigned rowStride = (unsigned)(H * C_TOTAL);

    // ---- Phase A: build weights + gather indices ----
    if (tid < 7) {                                    // wave0 lanes 0..6: Y axis
        const int p = tid;
        const float t  = (float)p * (1.0f / 6.0f);
        const float yc = (ry1 + t * (ry2 - ry1)) * sc;
        const float yf = floorf(yc);
        const float ly = yc - yf;
        const int yi0 = iclamp((int)yf, 0, H - 1);
        const int yi1 = iclamp((int)yf + 1, 0, H - 1);
        const float vy = (yc >= 0.0f && yc <= (float)(H - 1)) ? 1.0f : 0.0f;
        #pragma unroll
        for (int k = 0; k < 16; ++k) sWy[p][k] = 0.0f;
        sWy[p][2 * p]     = (1.0f - ly) * vy;
        sWy[p][2 * p + 1] = ly * vy;
        sRows[2 * p]     = yi0;
        sRows[2 * p + 1] = yi1;
        if (p == 6) { sRows[14] = yi1; sRows[15] = yi1; }   // clamped K padding
    }
    if (tid >= 32 && tid < 39) {                      // wave1 lanes 0..6: X axis
        const int p = tid - 32;
        const float t  = (float)p * (1.0f / 6.0f);
        const float xc = (rx1 + t * (rx2 - rx1)) * sc;
        const float xf = floorf(xc);
        const float lx = xc - xf;
        const int xi0 = iclamp((int)xf, 0, H - 1);
        const int xi1 = iclamp((int)xf + 1, 0, H - 1);
        const float vx = (xc >= 0.0f && xc <= (float)(H - 1)) ? 1.0f : 0.0f;
        #pragma unroll
        for (int k = 0; k < 16; ++k) sWx[p][k] = 0.0f;
        sWx[p][2 * p]     = (1.0f - lx) * vx;
        sWx[p][2 * p + 1] = lx * vx;
        sCols[2 * p]     = xi0;
        sCols[2 * p + 1] = xi1;
        if (p == 6) { sCols[14] = xi1; sCols[15] = xi1; }
    }
    __syncthreads();

    // ---- Per-lane WMMA A operands (16x4 f32 layout: m = lane&15, koff = (lane>>4)*2) ----
    const int n    = lane & 15;        // N (channel within block) for B/C/D
    const int m    = lane & 15;        // M for A operand
    const int koff = (lane >> 4) * 2;  // K sub-offset within a K=4 chunk

    v2f Ay[4], Ax[4];
    unsigned rowOff0[4], rowOff1[4];   // precomputed y-row element offsets (32-bit)
    #pragma unroll
    for (int kc = 0; kc < 4; ++kc) {
        if (m < 7) {
            Ay[kc].x = sWy[m][4 * kc + koff];
            Ay[kc].y = sWy[m][4 * kc + koff + 1];
            Ax[kc].x = sWx[m][4 * kc + koff];
            Ax[kc].y = sWx[m][4 * kc + koff + 1];
        } else {
            Ay[kc].x = 0.0f; Ay[kc].y = 0.0f;
            Ax[kc].x = 0.0f; Ax[kc].y = 0.0f;
        }
        rowOff0[kc] = (unsigned)sRows[4 * kc + koff]     * rowStride;
        rowOff1[kc] = (unsigned)sRows[4 * kc + koff + 1] * rowStride;
    }

    // ---- Main: each wave handles 2 channel blocks of 16 ----
    for (int cb = wave; cb < 16; cb += 8) {
        const int c0 = cb * 16;

        // Stage 1: contract over Y.  temp[py, xj, c] = Wy x G[:, xj, c-block]
        for (int xj = 0; xj < 14; ++xj) {
            const unsigned colOff = (unsigned)sCols[xj] * C_TOTAL + (unsigned)(c0 + n);
            v8f acc = {};
            #pragma unroll
            for (int kc = 0; kc < 4; ++kc) {
                v2f Bv;
                Bv.x = fm[rowOff0[kc] + colOff];
                Bv.y = fm[rowOff1[kc] + colOff];
                acc = __builtin_amdgcn_wmma_f32_16x16x4_f32(
                        false, Ay[kc], false, Bv, (short)0, acc, false, false);
            }
            if (lane < 16) {
                #pragma unroll
                for (int r = 0; r < 7; ++r)
                    sTemp[wave][r][xj][n] = acc[r];   // D: VGPR r -> M=r (lanes 0-15)
            }
        }
        __syncthreads();

        // Stage 2: contract over X.  out[py, px, c] = Wx x temp[py, :, c-block]
        for (int py = 0; py < 7; ++py) {
            v8f acc = {};
            #pragma unroll
            for (int kc = 0; kc < 4; ++kc) {
                int j0 = 4 * kc + koff;     if (j0 > 13) j0 = 13;  // zero-weight pad, valid addr
                int j1 = 4 * kc + koff + 1; if (j1 > 13) j1 = 13;
                v2f Bv;
                Bv.x = sTemp[wave][py][j0][n];
                Bv.y = sTemp[wave][py][j1][n];
                acc = __builtin_amdgcn_wmma_f32_16x16x4_f32(
                        false, Ax[kc], false, Bv, (short)0, acc, false, false);
            }
            if (lane < 16) {
                // Non-temporal: keep the 170MB pyramid resident in the 192MB L2;
                // the 51MB output is write-once streaming traffic.
                float* op = out + ((size_t)(b * nroi_per_img + roi) * 49 + py * 7) * C_TOTAL
                                + c0 + n;
                #pragma unroll
                for (int r = 0; r < 7; ++r)          // D row r == output px=r
                    __builtin_nontemporal_store(acc[r], op + (size_t)r * C_TOTAL);
            }
        }
        __syncthreads();   // sTemp reused by next channel block
    }
}

extern "C" void kernel_launch(void* const* d_in, const int* in_sizes, int n_in,
                              void* d_out, int out_size, void* d_ws, size_t ws_size,
                              hipStream_t stream) {
    (void)n_in; (void)out_size; (void)d_ws; (void)ws_size;
    const float* fm2  = (const float*)d_in[0];
    const float* fm3  = (const float*)d_in[1];
    const float* fm4  = (const float*)d_in[2];
    const float* fm5  = (const float*)d_in[3];
    const float* rois = (const float*)d_in[4];
    float* out = (float*)d_out;

    const int total_rois = in_sizes[4] / 4;              // B * N
    const int B = in_sizes[0] / (256 * 256 * C_TOTAL);   // fm2 = [B,256,256,256]
    const int N = total_rois / (B > 0 ? B : 1);

    roi_align_wmma_f32<<<dim3(total_rois), dim3(256), 0, stream>>>(
        fm2, fm3, fm4, fm5, rois, out, N);
}